// S4Layer_41137196761221
// MI455X (gfx1250) — compile-verified
//
#include <hip/hip_runtime.h>
#include <hip/hip_bf16.h>
#include <math.h>

typedef float v2f __attribute__((ext_vector_type(2)));
typedef float v8f __attribute__((ext_vector_type(8)));

constexpr int   Bb    = 8;
constexpr int   Ll    = 8192;
constexpr int   Dd    = 1024;
constexpr int   Nn    = 64;
constexpr int   TAPS  = 1024;       // e^{-0.05*1024} ~ 6e-23 << fp32 ulp -> exact truncation
constexpr int   DBLK  = TAPS / 16;  // 64 Toeplitz block-diagonals
constexpr float STEPf  = 0.1f;
constexpr float LN_EPSf = 1e-5f;

// ---------------------------------------------------------------------------
// Kernel 1: u[b,l] = dot(x[b,l,:], Wi) + bi      (one wave32 per token row)
// ---------------------------------------------------------------------------
__global__ void k_inproj(const float* __restrict__ x,
                         const float* __restrict__ Wi,
                         const float* __restrict__ bi,
                         float* __restrict__ u /* [L][8] */) {
  const int wave = threadIdx.x >> 5;
  const int lane = threadIdx.x & 31;
  const int row  = blockIdx.x * 8 + wave;            // row = b*L + l
  const float4* x4 = (const float4*)x + (size_t)row * (Dd / 4);
  const float4* w4 = (const float4*)Wi;
  float acc = 0.f;
#pragma unroll
  for (int q = 0; q < Dd / 4 / 32; ++q) {            // 8 x float4 per lane
    float4 a = x4[lane + 32 * q];
    float4 w = w4[lane + 32 * q];
    acc += a.x * w.x + a.y * w.y + a.z * w.z + a.w * w.w;
  }
#pragma unroll
  for (int m = 16; m >= 1; m >>= 1) acc += __shfl_xor(acc, m, 32);
  if (lane == 0) {
    const int b = row / Ll, l = row % Ll;
    u[l * 8 + b] = acc + bi[0];
  }
}

// ---------------------------------------------------------------------------
// Kernel 2: truncated S4D-ZOH kernel taps K[0..TAPS)
//   Ct_n = C_n * (exp(dt*A_n) - 1) / A_n ;  K[l] = 2*Re( sum_n Ct_n e^{dt A_n l} )
// ---------------------------------------------------------------------------
__global__ void k_kernelK(const float* __restrict__ A_re,
                          const float* __restrict__ A_im,
                          const float* __restrict__ C,
                          float* __restrict__ Kg /* [TAPS] */) {
  __shared__ float ctr[Nn], cti[Nn], dre[Nn], dim_[Nn];
  const int t = threadIdx.x;
  if (t < Nn) {
    const float ar = A_re[t], ai = A_im[t];
    const float tr = ar * STEPf, ti = ai * STEPf;    // dt*A
    const float decay = expf(tr);
    const float nr = decay * cosf(ti) - 1.f;         // exp(dtA) - 1
    const float ni = decay * sinf(ti);
    const float inv = 1.f / (ar * ar + ai * ai);
    const float qr = (nr * ar + ni * ai) * inv;      // (exp(dtA)-1)/A
    const float qi = (ni * ar - nr * ai) * inv;
    ctr[t] = C[t] * qr;  cti[t] = C[t] * qi;
    dre[t] = tr;         dim_[t] = ti;
  }
  __syncthreads();
  for (int l = t; l < TAPS; l += blockDim.x) {
    float acc = 0.f;
    for (int n = 0; n < Nn; ++n) {
      const double ph = (double)dim_[n] * (double)l; // large phase: keep in f64
      const float  dc = expf(dre[n] * (float)l);
      const float  cs = (float)cos(ph);
      const float  sn = (float)sin(ph);
      acc += dc * (ctr[n] * cs - cti[n] * sn);       // Re(Ct * e^{dtA l})
    }
    Kg[l] = 2.f * acc;
  }
}

// ---------------------------------------------------------------------------
// Kernel 3: y = T @ U via banded Toeplitz 16x16 blocks, V_WMMA_F32_16X16X4_F32
//   U is [L][16]: columns 0..7 = batches, 8..15 = zero (via cndmask on load).
//   One wave per 16-row output block i; accumulate over diagonals d=0..min(i,63).
// ---------------------------------------------------------------------------
__global__ void k_conv_wmma(const float* __restrict__ Kg,
                            const float* __restrict__ u,  /* [L][8] */
                            float* __restrict__ y)        /* [L][8] */ {
  __shared__ float Ks[TAPS];
  for (int t = threadIdx.x; t < TAPS; t += blockDim.x) Ks[t] = Kg[t];
  __syncthreads();

  const int wave = threadIdx.x >> 5;
  const int lane = threadIdx.x & 31;
  const int i    = blockIdx.x * 8 + wave;      // output block row (wave-uniform)
  const int m    = lane & 15;                  // A-matrix row held by this lane
  const int kb   = (lane >= 16) ? 2 : 0;       // K-dim pair {0,1} vs {2,3}
  const int n    = lane & 15;                  // B/C/D column (batch) index

  v8f c = {0.f, 0.f, 0.f, 0.f, 0.f, 0.f, 0.f, 0.f};
  const int dmax = (i < DBLK - 1) ? i : (DBLK - 1);

  for (int d = 0; d <= dmax; ++d) {
    const int jrow = (i - d) * 16;             // RHS block base row
#pragma unroll
    for (int s = 0; s < 4; ++s) {              // 16x16x16 = 4 x (16x16x4)
      // A_d[r][c] = K[16d + r - c], c = 4s + kb + {0,1}
      const int ia0 = 16 * d + m - (4 * s + kb);
      v2f a;
      a.x = (ia0 >= 0)     ? Ks[ia0]     : 0.f;
      a.y = (ia0 - 1 >= 0) ? Ks[ia0 - 1] : 0.f;
      // B[k][n] = U[jrow + 4s + kb + {0,1}][n]  (zero-padded columns n>=8)
      const int r0 = jrow + 4 * s + kb;
      v2f bv;
      bv.x = (n < 8) ? u[r0 * 8 + n]       : 0.f;
      bv.y = (n < 8) ? u[(r0 + 1) * 8 + n] : 0.f;
      c = __builtin_amdgcn_wmma_f32_16x16x4_f32(false, a, false, bv,
                                                (short)0, c, false, false);
    }
  }
  // C/D layout: VGPR v -> row v (lanes 0-15) / row v+8 (lanes 16-31), col = n
  if (n < 8) {
    const int rbase = i * 16 + ((lane >= 16) ? 8 : 0);
#pragma unroll
    for (int v = 0; v < 8; ++v) y[(rbase + v) * 8 + n] = c[v];
  }
}

// ---------------------------------------------------------------------------
// Kernel 4: out = LayerNorm(x + (y + Dskip*u) * Wo + bo)   (one wave per row,
// h kept in registers: single read of x, single write of out)
// ---------------------------------------------------------------------------
__global__ void k_out_ln(const float* __restrict__ x,
                         const float* __restrict__ u,
                         const float* __restrict__ yc,
                         const float* __restrict__ Dskip,
                         const float* __restrict__ Wo,
                         const float* __restrict__ bo,
                         const float* __restrict__ gamma,
                         const float* __restrict__ beta,
                         float* __restrict__ out) {
  const int wave = threadIdx.x >> 5;
  const int lane = threadIdx.x & 31;
  const int row  = blockIdx.x * 8 + wave;      // row = b*L + l
  const int b = row / Ll, l = row % Ll;
  const float ys = yc[l * 8 + b] + Dskip[0] * u[l * 8 + b];

  const float4* x4  = (const float4*)x  + (size_t)row * (Dd / 4);
  const float4* wo4 = (const float4*)Wo;
  const float4* bo4 = (const float4*)bo;

  float4 h[8];
  float s = 0.f, s2 = 0.f;
#pragma unroll
  for (int q = 0; q < 8; ++q) {
    const float4 xv = x4[lane + 32 * q];
    const float4 w  = wo4[lane + 32 * q];
    const float4 bb = bo4[lane + 32 * q];
    float4 hv;
    hv.x = xv.x + ys * w.x + bb.x;
    hv.y = xv.y + ys * w.y + bb.y;
    hv.z = xv.z + ys * w.z + bb.z;
    hv.w = xv.w + ys * w.w + bb.w;
    h[q] = hv;
    s  += hv.x + hv.y + hv.z + hv.w;
    s2 += hv.x * hv.x + hv.y * hv.y + hv.z * hv.z + hv.w * hv.w;
  }
#pragma unroll
  for (int mm = 16; mm >= 1; mm >>= 1) {
    s  += __shfl_xor(s,  mm, 32);
    s2 += __shfl_xor(s2, mm, 32);
  }
  const float mu   = s / (float)Dd;
  const float var  = s2 / (float)Dd - mu * mu;
  const float rinv = rsqrtf(var + LN_EPSf);

  float4* o4 = (float4*)out + (size_t)row * (Dd / 4);
  const float4* g4 = (const float4*)gamma;
  const float4* be = (const float4*)beta;
#pragma unroll
  for (int q = 0; q < 8; ++q) {
    const float4 g  = g4[lane + 32 * q];
    const float4 bt = be[lane + 32 * q];
    float4 o;
    o.x = g.x * (h[q].x - mu) * rinv + bt.x;
    o.y = g.y * (h[q].y - mu) * rinv + bt.y;
    o.z = g.z * (h[q].z - mu) * rinv + bt.z;
    o.w = g.w * (h[q].w - mu) * rinv + bt.w;
    o4[lane + 32 * q] = o;
  }
}

// ---------------------------------------------------------------------------
extern "C" void kernel_launch(void* const* d_in, const int* in_sizes, int n_in,
                              void* d_out, int out_size, void* d_ws, size_t ws_size,
                              hipStream_t stream) {
  const float* x     = (const float*)d_in[0];
  const float* A_re  = (const float*)d_in[1];
  const float* A_im  = (const float*)d_in[2];
  const float* C     = (const float*)d_in[3];
  const float* Dskip = (const float*)d_in[4];
  const float* Wi    = (const float*)d_in[5];
  const float* bi    = (const float*)d_in[6];
  const float* Wo    = (const float*)d_in[7];
  const float* bo    = (const float*)d_in[8];
  const float* gamma = (const float*)d_in[9];
  const float* beta  = (const float*)d_in[10];
  float* out = (float*)d_out;

  float* ws_u = (float*)d_ws;            // [L*8]
  float* ws_y = ws_u + (size_t)Ll * 8;   // [L*8]
  float* ws_K = ws_y + (size_t)Ll * 8;   // [TAPS]

  const int rows = Bb * Ll;              // 65536 token rows

  k_inproj<<<rows / 8, 256, 0, stream>>>(x, Wi, bi, ws_u);
  k_kernelK<<<1, 256, 0, stream>>>(A_re, A_im, C, ws_K);
  k_conv_wmma<<<(Ll / 16) / 8, 256, 0, stream>>>(ws_K, ws_u, ws_y);
  k_out_ln<<<rows / 8, 256, 0, stream>>>(x, ws_u, ws_y, Dskip, Wo, bo,
                                         gamma, beta, out);
}